// Quantizer_24043226923205
// MI455X (gfx1250) — compile-verified
//
#include <hip/hip_runtime.h>

typedef float v2f __attribute__((ext_vector_type(2)));
typedef float v8f __attribute__((ext_vector_type(8)));

#define VCODES 8192
#define CDIM   32
#define BATCH  8
#define NTOK   1024
#define TOKENS_PER_B 1360   // 16+64+256+1024

// ---------------------------------------------------------------------------
// Kernel 0: normalize codebook rows:  cbn[v] = cb[v] / (sum(cb[v]^2) + 1e-6)
// One thread per code; 32 floats = 8 float4 per code.
// ---------------------------------------------------------------------------
__global__ __launch_bounds__(256) void normalize_cb_kernel(
    const float* __restrict__ cb, float* __restrict__ cbn) {
  int v = blockIdx.x * blockDim.x + threadIdx.x;
  if (v >= VCODES) return;
  const float4* src = (const float4*)(cb + (size_t)v * CDIM);
  float4 r[8];
  float ss = 0.f;
#pragma unroll
  for (int i = 0; i < 8; ++i) {
    r[i] = src[i];
    ss += r[i].x * r[i].x + r[i].y * r[i].y + r[i].z * r[i].z + r[i].w * r[i].w;
  }
  float inv = 1.0f / (ss + 1e-6f);
  float4* dst = (float4*)(cbn + (size_t)v * CDIM);
#pragma unroll
  for (int i = 0; i < 8; ++i) {
    float4 t = r[i];
    t.x *= inv; t.y *= inv; t.z *= inv; t.w *= inv;
    dst[i] = t;
  }
}

// ---------------------------------------------------------------------------
// Main kernel: per workgroup, 16 rows of F (normalized into LDS) vs all 8192
// codes via V_WMMA_F32_16X16X4_F32 chains; argmax reduction; scatter to the
// 4 concatenated scales.
//   grid = 8192/16 = 512 workgroups, 256 threads (8 wave32).
// ---------------------------------------------------------------------------
__global__ __launch_bounds__(256) void quantize_kernel(
    const float* __restrict__ f, const float* __restrict__ cbn,
    float* __restrict__ out) {
  __shared__ float sF[16 * 33];          // padded stride 33 -> no LDS bank conflicts
  __shared__ float sMax[8][16];
  __shared__ int   sIdx[8][16];

  const int tid  = threadIdx.x;
  const int row0 = blockIdx.x * 16;      // global row tile (0..8191 in steps of 16)

  // --- normalize 16 F rows into LDS (threads 0..15, one row each) ---
  if (tid < 16) {
    const float4* src = (const float4*)(f + (size_t)(row0 + tid) * CDIM);
    float4 r[8];
    float ss = 0.f;
#pragma unroll
    for (int i = 0; i < 8; ++i) {
      r[i] = src[i];
      ss += r[i].x * r[i].x + r[i].y * r[i].y + r[i].z * r[i].z + r[i].w * r[i].w;
    }
    float inv = 1.0f / (ss + 1e-6f);
    float* dst = sF + tid * 33;
#pragma unroll
    for (int i = 0; i < 8; ++i) {
      dst[4 * i + 0] = r[i].x * inv;
      dst[4 * i + 1] = r[i].y * inv;
      dst[4 * i + 2] = r[i].z * inv;
      dst[4 * i + 3] = r[i].w * inv;
    }
  }
  __syncthreads();

  const int wave = tid >> 5;             // 0..7
  const int lane = tid & 31;
  const int hf   = lane >> 4;            // lane half (0/1)
  const int lj   = lane & 15;            // position within half

  // --- A fragments (constant across all column passes) ---
  // ISA 32-bit A 16x4 layout: lane l, VGPR v -> M = l%16, K = v + 2*(l/16)
  v2f a[8];
#pragma unroll
  for (int k = 0; k < 8; ++k) {
    const float* p = sF + lj * 33 + 4 * k + 2 * hf;
    a[k][0] = p[0];
    a[k][1] = p[1];
  }

  float bestV[8];
  int   bestI[8];
#pragma unroll
  for (int r = 0; r < 8; ++r) { bestV[r] = -3.4e38f; bestI[r] = 0x7fffffff; }

  // --- sweep all 8192 codes: 8 waves x 16 cols = 128 cols/pass, 64 passes ---
  for (int pass = 0; pass < VCODES / 128; ++pass) {
    const int col = pass * 128 + wave * 16 + lj;
    // B 4x16 layout (mirror of A): lane l, VGPR v -> N = l%16, K = v + 2*(l/16)
    // B[k][n] = cbn[n][k]  (B = Cbn^T)
    const float* cb = cbn + (size_t)col * CDIM + 2 * hf;
    v2f b[8];
#pragma unroll
    for (int k = 0; k < 8; ++k) {
      b[k][0] = cb[4 * k + 0];
      b[k][1] = cb[4 * k + 1];
    }
    v8f c = {};
#pragma unroll
    for (int k = 0; k < 8; ++k) {
      // 8 args: (neg_a, A, neg_b, B, c_mod, C, reuse_a, reuse_b)
      c = __builtin_amdgcn_wmma_f32_16x16x4_f32(
          false, a[k], false, b[k], (short)0, c, false, false);
    }
    // D layout: lane l, VGPR r -> row = r + 8*(l/16), col = col
#pragma unroll
    for (int r = 0; r < 8; ++r) {
      float v = c[r];
      if (v > bestV[r] || (v == bestV[r] && col < bestI[r])) {
        bestV[r] = v; bestI[r] = col;
      }
    }
  }

  // --- reduce across the 16 columns held in each 16-lane half ---
#pragma unroll
  for (int m = 1; m < 16; m <<= 1) {
#pragma unroll
    for (int r = 0; r < 8; ++r) {
      float ov = __shfl_xor(bestV[r], m, 32);
      int   oi = __shfl_xor(bestI[r], m, 32);
      if (ov > bestV[r] || (ov == bestV[r] && oi < bestI[r])) {
        bestV[r] = ov; bestI[r] = oi;
      }
    }
  }
  if (lj == 0) {
#pragma unroll
    for (int r = 0; r < 8; ++r) {
      sMax[wave][hf * 8 + r] = bestV[r];
      sIdx[wave][hf * 8 + r] = bestI[r];
    }
  }
  __syncthreads();

  // --- final cross-wave reduction + scatter (threads 0..15, one row each) ---
  if (tid < 16) {
    float bv = sMax[0][tid];
    int   bi = sIdx[0][tid];
#pragma unroll
    for (int w = 1; w < 8; ++w) {
      float v = sMax[w][tid];
      int   i = sIdx[w][tid];
      if (v > bv || (v == bv && i < bi)) { bv = v; bi = i; }
    }
    const int g  = row0 + tid;          // global row 0..8191
    const int bb = g >> 10;             // batch
    const int n  = g & 1023;            // token index within batch
    const int valsOff = BATCH * TOKENS_PER_B;
    const float fbi = (float)bi;        // indices <= 8191, exact in f32

    // pn=1024 (base 336): every row
    int o = bb * TOKENS_PER_B + 336 + n;
    out[o] = fbi;  out[valsOff + o] = bv;
    // pn=256 (base 80): rows n = 4i+2
    if ((n & 3) == 2) {
      int o2 = bb * TOKENS_PER_B + 80 + ((n - 2) >> 2);
      out[o2] = fbi;  out[valsOff + o2] = bv;
    }
    // pn=64 (base 16): rows n = 16i+8
    if ((n & 15) == 8) {
      int o3 = bb * TOKENS_PER_B + 16 + ((n - 8) >> 4);
      out[o3] = fbi;  out[valsOff + o3] = bv;
    }
    // pn=16 (base 0): rows n = 64i+32
    if ((n & 63) == 32) {
      int o4 = bb * TOKENS_PER_B + ((n - 32) >> 6);
      out[o4] = fbi;  out[valsOff + o4] = bv;
    }
  }
}

// ---------------------------------------------------------------------------
extern "C" void kernel_launch(void* const* d_in, const int* in_sizes, int n_in,
                              void* d_out, int out_size, void* d_ws, size_t ws_size,
                              hipStream_t stream) {
  (void)in_sizes; (void)n_in; (void)out_size; (void)ws_size;
  const float* f  = (const float*)d_in[0];   // (8,1024,32) f32
  const float* cb = (const float*)d_in[1];   // (8192,32)  f32
  float* out = (float*)d_out;                // [8*1360 idx-as-float | 8*1360 vals]
  float* cbn = (float*)d_ws;                 // 8192*32 f32 = 1 MB scratch

  normalize_cb_kernel<<<VCODES / 256, 256, 0, stream>>>(cb, cbn);
  quantize_kernel<<<(BATCH * NTOK) / 16, 256, 0, stream>>>(f, cbn, out);
}